// LiDARSemanticFeatureAggregationModule_69750268887275
// MI455X (gfx1250) — compile-verified
//
#include <hip/hip_runtime.h>

typedef __attribute__((ext_vector_type(2))) float v2f;
typedef __attribute__((ext_vector_type(8))) float v8f;

#define KCLS  20   // classes
#define CCH   64   // feature channels
#define BSEG  4    // batch segments
#define ACC_K 32   // classes padded to 2 WMMA M-tiles

// ---------------- workspace layout (floats) ----------------
// [0,80)    : mkeys (unsigned, order-preserving float keys)
// [80,160)  : m   (segment max, decoded)
// [160,240) : s   (segment sum of exp)
// [240,244) : seg_lo (int)
// [244,248) : seg_hi (int)
// [256,256+BSEG*ACC_K*CCH) : acc  [B][32][64]

__global__ void lidar_init_ws(unsigned* mkeys, float* s, int* slo, int* shi,
                              float* acc, int accn, int N) {
    int t = blockIdx.x * blockDim.x + threadIdx.x;
    if (t < BSEG * KCLS) { mkeys[t] = 0u; s[t] = 0.f; }
    if (t < BSEG) { slo[t] = N; shi[t] = 0; }
    if (t < accn) acc[t] = 0.f;
}

// batch_idx is sorted -> segments are contiguous ranges; detect boundaries
__global__ void lidar_seg_bounds(const int* __restrict__ bidx,
                                 int* __restrict__ slo, int* __restrict__ shi, int N) {
    int stride = gridDim.x * blockDim.x;
    for (int t = blockIdx.x * blockDim.x + threadIdx.x; t < N; t += stride) {
        int b = bidx[t];
        if (t == 0 || bidx[t - 1] != b) atomicMin(&slo[b], t);
        if (t == N - 1 || bidx[t + 1] != b) atomicMax(&shi[b], t + 1);
    }
}

// segment max via order-preserving u32 keys; LDS reduction first
__global__ void lidar_seg_max(const float* __restrict__ probs,
                              const int* __restrict__ bidx,
                              unsigned* __restrict__ mkeys, int N) {
    __shared__ unsigned lmax[BSEG * KCLS];
    for (int t = threadIdx.x; t < BSEG * KCLS; t += blockDim.x) lmax[t] = 0u;
    __syncthreads();
    int total = N * KCLS;
    int stride = gridDim.x * blockDim.x;
    for (int t = blockIdx.x * blockDim.x + threadIdx.x; t < total; t += stride) {
        int i = t / KCLS;
        int j = t - i * KCLS;
        int b = bidx[i];
        unsigned bits = __float_as_uint(probs[t]);
        unsigned key = (bits & 0x80000000u) ? ~bits : (bits | 0x80000000u);
        atomicMax(&lmax[b * KCLS + j], key);
    }
    __syncthreads();
    for (int t = threadIdx.x; t < BSEG * KCLS; t += blockDim.x)
        atomicMax(&mkeys[t], lmax[t]);
}

__global__ void lidar_decode_max(const unsigned* __restrict__ mkeys, float* __restrict__ m) {
    int t = blockIdx.x * blockDim.x + threadIdx.x;
    if (t < BSEG * KCLS) {
        unsigned u = mkeys[t];
        unsigned bits = (u & 0x80000000u) ? (u & 0x7FFFFFFFu) : ~u;
        m[t] = __uint_as_float(bits);
    }
}

__global__ void lidar_seg_sum(const float* __restrict__ probs,
                              const int* __restrict__ bidx,
                              const float* __restrict__ m,
                              float* __restrict__ s, int N) {
    __shared__ float lsum[BSEG * KCLS];
    for (int t = threadIdx.x; t < BSEG * KCLS; t += blockDim.x) lsum[t] = 0.f;
    __syncthreads();
    int total = N * KCLS;
    int stride = gridDim.x * blockDim.x;
    for (int t = blockIdx.x * blockDim.x + threadIdx.x; t < total; t += stride) {
        int i = t / KCLS;
        int j = t - i * KCLS;
        int b = bidx[i];
        float e = __expf(probs[t] - m[b * KCLS + j]);
        atomicAdd(&lsum[b * KCLS + j], e);
    }
    __syncthreads();
    for (int t = threadIdx.x; t < BSEG * KCLS; t += blockDim.x)
        atomicAdd(&s[t], lsum[t]);
}

// Main kernel: per-wave chunk intersected with each contiguous segment range.
// All loop bounds forced wave-uniform (readfirstlane) -> scalar loop control,
// EXEC guaranteed all-ones at every WMMA. Steady loop: unconditional coalesced
// loads + 8x v_wmma_f32_16x16x4_f32 per 4-point group, unrolled x2.
__global__ void lidar_wmma_aggregate(const float* __restrict__ feats,
                                     const float* __restrict__ probs,
                                     const float* __restrict__ m,
                                     const float* __restrict__ s,
                                     const int* __restrict__ slo,
                                     const int* __restrict__ shi,
                                     float* __restrict__ acc,
                                     int N, int chunk) {
    __shared__ float sm[BSEG * KCLS];
    __shared__ float srs[BSEG * KCLS];
    for (int t = threadIdx.x; t < BSEG * KCLS; t += blockDim.x) {
        sm[t] = m[t];
        srs[t] = 1.0f / s[t];
    }
    __syncthreads();

    int wave = (blockIdx.x * blockDim.x + threadIdx.x) >> 5;
    int lane = threadIdx.x & 31;
    int start = __builtin_amdgcn_readfirstlane(wave * chunk);  // wave-uniform scalar
    if (start >= N) return;                 // whole wave exits together
    int end = min(N, start + chunk);

    int half = lane >> 4;                   // 0: K=0,1 ; 1: K=2,3 (A and B layouts)
    int lid  = lane & 15;                   // A row (class), B column (channel)
    int cls1  = 16 + lid;                   // tile-1 class for this lane
    bool has1 = (cls1 < KCLS);              // only lid<4 real
    int cls1c = has1 ? cls1 : (KCLS - 1);   // clamped for safe addressing

    for (int b = 0; b < BSEG; ++b) {
        int lo = __builtin_amdgcn_readfirstlane(max(start, slo[b]));  // scalar
        int hi = __builtin_amdgcn_readfirstlane(min(end, shi[b]));    // scalar
        if (lo >= hi) continue;             // scalar branch, EXEC stays full

        float m0 = sm[b * KCLS + lid];
        float r0 = srs[b * KCLS + lid];
        float m1 = sm[b * KCLS + cls1c];
        float r1 = srs[b * KCLS + cls1c];

        v8f c[2][4];
        #pragma unroll
        for (int mt = 0; mt < 2; ++mt)
            #pragma unroll
            for (int nt = 0; nt < 4; ++nt)
                c[mt][nt] = (v8f){0.f, 0.f, 0.f, 0.f, 0.f, 0.f, 0.f, 0.f};

        int g = lo;
        // ---------------- steady loop: no guards, no clamps ----------------
        #pragma unroll 2
        for (; g + 4 <= hi; g += 4) {
            int p0 = g + half * 2;
            int p1 = p0 + 1;

            float t00 = probs[p0 * KCLS + lid];
            float t01 = probs[p1 * KCLS + lid];
            float t10 = probs[p0 * KCLS + cls1c];
            float t11 = probs[p1 * KCLS + cls1c];

            float e10 = __expf(t10 - m1) * r1;
            float e11 = __expf(t11 - m1) * r1;
            v2f a0 = {__expf(t00 - m0) * r0, __expf(t01 - m0) * r0};
            v2f a1 = {has1 ? e10 : 0.f, has1 ? e11 : 0.f};

            const float* f0 = feats + (size_t)p0 * CCH + lid;
            const float* f1 = feats + (size_t)p1 * CCH + lid;
            v2f bt[4];
            bt[0] = (v2f){f0[0],  f1[0]};
            bt[1] = (v2f){f0[16], f1[16]};
            bt[2] = (v2f){f0[32], f1[32]};
            bt[3] = (v2f){f0[48], f1[48]};

            #pragma unroll
            for (int nt = 0; nt < 4; ++nt) {
                c[0][nt] = __builtin_amdgcn_wmma_f32_16x16x4_f32(
                    false, a0, false, bt[nt], (short)0, c[0][nt], false, false);
                c[1][nt] = __builtin_amdgcn_wmma_f32_16x16x4_f32(
                    false, a1, false, bt[nt], (short)0, c[1][nt], false, false);
            }
        }
        // ---------------- tail group (< 4 points), scalar-guarded ----------------
        if (g < hi) {
            int p0 = g + half * 2;
            int p1 = p0 + 1;
            int q0 = min(p0, hi - 1);
            int q1 = min(p1, hi - 1);
            bool v0 = p0 < hi;
            bool v1 = p1 < hi;

            float t00 = probs[q0 * KCLS + lid];
            float t01 = probs[q1 * KCLS + lid];
            float t10 = probs[q0 * KCLS + cls1c];
            float t11 = probs[q1 * KCLS + cls1c];

            float e00 = __expf(t00 - m0) * r0;
            float e01 = __expf(t01 - m0) * r0;
            float e10 = __expf(t10 - m1) * r1;
            float e11 = __expf(t11 - m1) * r1;
            v2f a0 = {v0 ? e00 : 0.f, v1 ? e01 : 0.f};
            v2f a1 = {(v0 && has1) ? e10 : 0.f, (v1 && has1) ? e11 : 0.f};

            const float* f0 = feats + (size_t)q0 * CCH + lid;
            const float* f1 = feats + (size_t)q1 * CCH + lid;
            v2f bt[4];
            bt[0] = (v2f){f0[0],  f1[0]};
            bt[1] = (v2f){f0[16], f1[16]};
            bt[2] = (v2f){f0[32], f1[32]};
            bt[3] = (v2f){f0[48], f1[48]};

            #pragma unroll
            for (int nt = 0; nt < 4; ++nt) {
                c[0][nt] = __builtin_amdgcn_wmma_f32_16x16x4_f32(
                    false, a0, false, bt[nt], (short)0, c[0][nt], false, false);
                c[1][nt] = __builtin_amdgcn_wmma_f32_16x16x4_f32(
                    false, a1, false, bt[nt], (short)0, c[1][nt], false, false);
            }
        }

        // finalize: C/D layout is VGPR r -> rows (half*8 + r); cols = nt*16 + lid
        #pragma unroll
        for (int nt = 0; nt < 4; ++nt) {
            int ch = nt * 16 + lid;
            #pragma unroll
            for (int r = 0; r < 8; ++r) {
                int k0 = half * 8 + r;
                atomicAdd(&acc[((b * ACC_K) + k0) * CCH + ch], c[0][nt][r]);
                int k1 = 16 + half * 8 + r;
                if (k1 < KCLS)
                    atomicAdd(&acc[((b * ACC_K) + k1) * CCH + ch], c[1][nt][r]);
            }
        }
    }
}

// [B][32][64] accumulator -> output [B][C][K][1]
__global__ void lidar_writeout(const float* __restrict__ acc, float* __restrict__ out) {
    int t = blockIdx.x * blockDim.x + threadIdx.x;
    if (t >= BSEG * CCH * KCLS) return;
    int kk = t % KCLS;
    int ch = (t / KCLS) % CCH;
    int b  = t / (KCLS * CCH);
    out[t] = acc[(b * ACC_K + kk) * CCH + ch];
}

extern "C" void kernel_launch(void* const* d_in, const int* in_sizes, int n_in,
                              void* d_out, int out_size, void* d_ws, size_t ws_size,
                              hipStream_t stream) {
    const float* feats = (const float*)d_in[0];
    const float* probs = (const float*)d_in[1];
    const int*   bidx  = (const int*)d_in[2];
    int N = in_sizes[2];

    float*    ws    = (float*)d_ws;
    unsigned* mkeys = (unsigned*)ws;         // 80
    float*    m     = ws + 80;               // 80
    float*    s     = ws + 160;              // 80
    int*      slo   = (int*)(ws + 240);      // 4
    int*      shi   = (int*)(ws + 244);      // 4
    float*    acc   = ws + 256;              // 4*32*64 = 8192
    int accn = BSEG * ACC_K * CCH;

    lidar_init_ws<<<(accn + 255) / 256, 256, 0, stream>>>(mkeys, s, slo, shi, acc, accn, N);
    lidar_seg_bounds<<<128, 256, 0, stream>>>(bidx, slo, shi, N);
    lidar_seg_max<<<256, 256, 0, stream>>>(probs, bidx, mkeys, N);
    lidar_decode_max<<<1, 128, 0, stream>>>(mkeys, m);
    lidar_seg_sum<<<256, 256, 0, stream>>>(probs, bidx, m, s, N);

    const int WAVES = 1024;                  // 128 blocks x 8 waves
    int chunk = ((N + WAVES - 1) / WAVES + 3) & ~3;
    lidar_wmma_aggregate<<<WAVES / 8, 256, 0, stream>>>(feats, probs, m, s, slo, shi, acc, N, chunk);

    lidar_writeout<<<(BSEG * CCH * KCLS + 255) / 256, 256, 0, stream>>>(acc, (float*)d_out);
}